// TestModel_82454782148650
// MI455X (gfx1250) — compile-verified
//
#include <hip/hip_runtime.h>

// ---------------------------------------------------------------------------
// Problem constants (match reference)
// ---------------------------------------------------------------------------
#define BATCH 8
#define NPTS  4096
#define KNN   16
#define MOUT  (NPTS / 2)          // 2048
#define NT    (NPTS / 16)         // 256 column tiles of 16

typedef __attribute__((ext_vector_type(2))) float v2f;
typedef __attribute__((ext_vector_type(8))) float v8f;

// ---------------------------------------------------------------------------
// Kernel 1: fused kNN feature. One wave per 16-row tile.
// Per loop iteration: TWO 16x16 Gram tiles via V_WMMA_F32_16X16X4_F32
// (A = 16x4 rows [x,y,z,0]), distances written as a 16x32 LDS strip.
// Lanes 0-15 run top-16 insertion for even tiles, lanes 16-31 for odd tiles
// (all 32 lanes busy); the two sorted half-lists are merged once at the end.
// Insertion is fully branchless (cmp + select, compile-time indices).
// ---------------------------------------------------------------------------
__global__ __launch_bounds__(32)
void knn_feats_kernel(const float* __restrict__ x, float* __restrict__ feats) {
    const int wid  = blockIdx.x;            // 0 .. BATCH*NT-1
    const int b    = wid / NT;
    const int rt   = wid % NT;
    const int row0 = rt * 16;

    const int lane = threadIdx.x;           // 0..31 (wave32)
    const int l16  = lane & 15;
    const int half = lane >> 4;             // A/B layout: 0 -> K={0,1}, 1 -> K={2,3}

    const float* xb = x + (size_t)b * NPTS * 3;

    __shared__ float d2r[16];
    __shared__ float Dt[16][33];            // 16 rows x 32 cols (+1 pad)
    __shared__ float sd[32][KNN];           // per-lane sorted top-16 dists
    __shared__ int   si[32][KNN];           // per-lane sorted top-16 indices

    // ---- A matrix: row = row0 + l16; lanes 0-15 hold K={0,1}=(x,y),
    //      lanes 16-31 hold K={2,3}=(z,0) of the same row.
    const int row = row0 + l16;
    const float rx = xb[row * 3 + 0];
    const float ry = xb[row * 3 + 1];
    const float rz = xb[row * 3 + 2];
    v2f A;
    A.x = half ? rz : rx;
    A.y = half ? 0.0f : ry;
    if (lane < 16) d2r[lane] = rx * rx + ry * ry + rz * rz;
    __syncthreads();

    float bestd[KNN];
    int   besti[KNN];
#pragma unroll
    for (int k = 0; k < KNN; ++k) { bestd[k] = 3.0e30f; besti[k] = 0; }

    for (int ct = 0; ct < NT; ct += 2) {
        const int col0 = ct * 16;

        if (ct + 2 < NT)  // WGP-scope prefetch of the next tile pair
            __builtin_prefetch(xb + (size_t)(col0 + 32) * 3, 0, 3);

        // ---- B matrices for column tiles ct and ct+1
        const int c0 = col0 + l16;
        const int c1 = col0 + 16 + l16;
        const float q0x = xb[c0 * 3 + 0], q0y = xb[c0 * 3 + 1], q0z = xb[c0 * 3 + 2];
        const float q1x = xb[c1 * 3 + 0], q1y = xb[c1 * 3 + 1], q1z = xb[c1 * 3 + 2];
        v2f B0, B1;
        B0.x = half ? q0z : q0x;  B0.y = half ? 0.0f : q0y;
        B1.x = half ? q1z : q1x;  B1.y = half ? 0.0f : q1y;
        const float d2c0 = q0x * q0x + q0y * q0y + q0z * q0z;
        const float d2c1 = q1x * q1x + q1y * q1y + q1z * q1z;

        // ---- two Gram tiles on the matrix pipe
        v8f C0 = {}, C1 = {};
        C0 = __builtin_amdgcn_wmma_f32_16x16x4_f32(false, A, false, B0,
                                                   (short)0, C0, false, false);
        C1 = __builtin_amdgcn_wmma_f32_16x16x4_f32(false, A, false, B1,
                                                   (short)0, C1, false, false);

        // C layout: VGPR r holds row (r + 8*half), column = l16
#pragma unroll
        for (int r = 0; r < 8; ++r) {
            const int m = r + half * 8;
            Dt[m][l16]      = d2r[m] + d2c0 - 2.0f * C0[r];
            Dt[m][16 + l16] = d2r[m] + d2c1 - 2.0f * C1[r];
        }
        __syncthreads();

        // ---- all 32 lanes insert: lane owns row l16, half selects the tile
        const int cb = col0 + half * 16;
#pragma unroll
        for (int j = 0; j < 16; ++j) {
            float cd = Dt[l16][half * 16 + j];
            int   ci = cb + j;
            if (cd < bestd[KNN - 1]) {      // cheap reject of most candidates
#pragma unroll
                for (int p = 0; p < KNN; ++p) {   // branchless bubble insert
                    const bool  c  = cd < bestd[p];
                    const float od = bestd[p];
                    const int   oi = besti[p];
                    bestd[p] = c ? cd : od;
                    besti[p] = c ? ci : oi;
                    cd = c ? od : cd;
                    ci = c ? oi : ci;
                }
            }
        }
        __syncthreads();
    }

    // ---- dump both half-lists, merge per row, average neighbor coords
#pragma unroll
    for (int k = 0; k < KNN; ++k) { sd[lane][k] = bestd[k]; si[lane][k] = besti[k]; }
    __syncthreads();

    if (lane < 16) {
        int ia = 0, ib = 0;
        float sx = 0.0f, sy = 0.0f, sz = 0.0f;
#pragma unroll
        for (int k = 0; k < KNN; ++k) {     // merge two sorted ascending lists
            const float da = sd[lane][ia];
            const float db = sd[lane + 16][ib];
            int idx;
            if (da <= db) { idx = si[lane][ia]; ++ia; }
            else          { idx = si[lane + 16][ib]; ++ib; }
            sx += xb[idx * 3 + 0];
            sy += xb[idx * 3 + 1];
            sz += xb[idx * 3 + 2];
        }
        float* f = feats + ((size_t)b * NPTS + row0 + lane) * 3;
        const float inv = 1.0f / (float)KNN;
        f[0] = sx * inv;
        f[1] = sy * inv;
        f[2] = sz * inv;
    }
}

// ---------------------------------------------------------------------------
// Kernel 2: farthest point sampling, one 256-thread workgroup per batch.
// Coords + running min-dist live in registers (16 points per thread).
// Argmax = wave32 __shfl_xor butterfly + 8-entry cross-wave LDS stage:
// only 2 barriers per step (vs 9 with an LDS tree).
// Tie-break = lowest index (matches jnp.argmax first-max semantics).
// ---------------------------------------------------------------------------
__global__ __launch_bounds__(256)
void fps_kernel(const float* __restrict__ x, int* __restrict__ sidx) {
    const int b   = blockIdx.x;
    const int tid = threadIdx.x;
    const int PPT = NPTS / 256;             // 16 points per thread
    const float* xb = x + (size_t)b * NPTS * 3;

    float px[16], py[16], pz[16], mind[16];
#pragma unroll
    for (int j = 0; j < PPT; ++j) {
        const int p = tid * PPT + j;
        px[j] = xb[p * 3 + 0];
        py[j] = xb[p * 3 + 1];
        pz[j] = xb[p * 3 + 2];
        mind[j] = 1.0e10f;
    }

    __shared__ float wval[8];
    __shared__ int   widx[8];
    __shared__ int   s_last;

    int last = 0;
    for (int i = 0; i < MOUT; ++i) {
        if (tid == 0) sidx[(size_t)b * MOUT + i] = last;

        const float lx = xb[last * 3 + 0];
        const float ly = xb[last * 3 + 1];
        const float lz = xb[last * 3 + 2];

        float lm = -1.0f;
        int   li = tid * PPT;
#pragma unroll
        for (int j = 0; j < PPT; ++j) {
            const float dx = px[j] - lx;
            const float dy = py[j] - ly;
            const float dz = pz[j] - lz;
            const float d  = dx * dx + dy * dy + dz * dz;
            mind[j] = fminf(mind[j], d);
            if (mind[j] > lm) { lm = mind[j]; li = tid * PPT + j; }
        }

        // wave32 butterfly argmax (ds_permute class ops, no barriers)
#pragma unroll
        for (int off = 16; off > 0; off >>= 1) {
            const float ov = __shfl_xor(lm, off, 32);
            const int   oi = __shfl_xor(li, off, 32);
            if (ov > lm || (ov == lm && oi < li)) { lm = ov; li = oi; }
        }
        if ((tid & 31) == 0) { wval[tid >> 5] = lm; widx[tid >> 5] = li; }
        __syncthreads();

        if (tid == 0) {
            float bm = wval[0]; int bi = widx[0];
#pragma unroll
            for (int w = 1; w < 8; ++w) {
                if (wval[w] > bm || (wval[w] == bm && widx[w] < bi)) {
                    bm = wval[w]; bi = widx[w];
                }
            }
            s_last = bi;
        }
        __syncthreads();
        last = s_last;
    }
}

// ---------------------------------------------------------------------------
// Kernel 3: gather feats at FPS indices -> out [B, M, 3]
// ---------------------------------------------------------------------------
__global__ __launch_bounds__(256)
void gather_kernel(const float* __restrict__ feats, const int* __restrict__ sidx,
                   float* __restrict__ out) {
    const int t = blockIdx.x * blockDim.x + threadIdx.x;
    if (t >= BATCH * MOUT) return;
    const int b = t / MOUT;
    const int s = sidx[t];
    const float* f = feats + ((size_t)b * NPTS + s) * 3;
    out[(size_t)t * 3 + 0] = f[0];
    out[(size_t)t * 3 + 1] = f[1];
    out[(size_t)t * 3 + 2] = f[2];
}

// ---------------------------------------------------------------------------
extern "C" void kernel_launch(void* const* d_in, const int* in_sizes, int n_in,
                              void* d_out, int out_size, void* d_ws, size_t ws_size,
                              hipStream_t stream) {
    const float* x = (const float*)d_in[0];           // [8, 4096, 3] fp32
    float* feats = (float*)d_ws;                       // B*N*3 floats = 393216 B
    int*   sidx  = (int*)((char*)d_ws + (size_t)BATCH * NPTS * 3 * sizeof(float));
    float* out   = (float*)d_out;                      // [8, 2048, 3]

    knn_feats_kernel<<<BATCH * NT, 32, 0, stream>>>(x, feats);
    fps_kernel<<<BATCH, 256, 0, stream>>>(x, sidx);
    const int tot = BATCH * MOUT;
    gather_kernel<<<(tot + 255) / 256, 256, 0, stream>>>(feats, sidx, out);
}